// LangevinApproxDRO_403726926609
// MI455X (gfx1250) — compile-verified
//
#include <hip/hip_runtime.h>
#include <math.h>

// ---------------- problem constants (from reference) ----------------
#define T_STEPS 1000
#define M_KEEP  100
#define LR_     0.1f
#define SIGMA_  1.0f
#define SIGMA_L 1.0f
#define EPS_    1.0f
#define RHO_    0.1f
#define BOUND_  100.0f
#define B_ROWS  4096
#define DIN_    127
#define DD_     128

// ---------------- launch shape ----------------
#define NB   64            // persistent blocks (grid barrier spans these)
#define TPB  128           // 4 waves / block (wave32)
#define WPB  (TPB / 32)
#define ROWS_PER_WAVE 16   // one WMMA 16x128 tile per wave; NB*WPB*16 == 4096

typedef __attribute__((ext_vector_type(2))) float v2f;
typedef __attribute__((ext_vector_type(8))) float v8f;

// ---------------- counter-based RNG (Box-Muller) ----------------
__device__ __forceinline__ unsigned hash_u32(unsigned v) {
    v ^= 2747636419u; v *= 2654435769u;
    v ^= v >> 16;     v *= 2654435769u;
    v ^= v >> 16;     v *= 2654435769u;
    return v;
}

__device__ __forceinline__ float gauss_rand(unsigned id, unsigned t) {
    unsigned s0 = id * 0x9E3779B9u + t * 0x85EBCA6Bu + 0x01234567u;
    unsigned h1 = hash_u32(s0);
    unsigned h2 = hash_u32(s0 ^ 0xC2B2AE35u);
    float u1 = ((h1 >> 9) + 1u) * (1.0f / 8388608.0f);   // (0, 1]
    float u2 = (h2 >> 9) * (1.0f / 8388608.0f);          // [0, 1)
    float rr = __builtin_sqrtf(-2.0f * __logf(u1));
    return rr * __cosf(6.28318530718f * u2);
}

// ---------------- grid-wide barrier (persistent kernel) ----------------
__device__ __forceinline__ void grid_barrier(volatile unsigned* cnt,
                                             volatile unsigned* gen,
                                             unsigned nblocks) {
    __syncthreads();
    if (threadIdx.x == 0) {
        __threadfence();
        unsigned g = *gen;
        unsigned prev = atomicAdd((unsigned*)cnt, 1u);
        if (prev == nblocks - 1u) {
            *cnt = 0u;
            __threadfence();
            atomicAdd((unsigned*)gen, 1u);
        } else {
            while (*gen == g) { __builtin_amdgcn_s_sleep(2); }
        }
    }
    __syncthreads();
    __threadfence();
}

// ---------------- persistent Langevin-DRO kernel ----------------
__global__ __launch_bounds__(TPB, 1)
void langevin_dro_kernel(const float* __restrict__ x, const float* __restrict__ y,
                         const float* __restrict__ w, const float* __restrict__ b0,
                         const float* __restrict__ lbd, float* __restrict__ out,
                         unsigned* __restrict__ bar, float* __restrict__ normP,
                         float* __restrict__ lossP) {
    __shared__ float sred[WPB];

    const int lane   = threadIdx.x & 31;
    const int wInB   = threadIdx.x >> 5;
    const int waveId = blockIdx.x * WPB + wInB;
    const int myRow  = waveId * ROWS_PER_WAVE + (lane & 15);
    const int h      = lane >> 4;           // lane half selects cols {0,1} vs {2,3} mod 4

    const float b0c  = b0[0];
    const float lbdc = lbd[0];
    const float c2   = lbdc + EPS_ / (SIGMA_ * SIGMA_);   // (lbd + eps/sigma^2)
    const float nsc  = __builtin_sqrtf(2.0f * LR_) * SIGMA_L;

    // Per-lane slice of the wave's 16x128 tile, stored directly in the
    // V_WMMA_F32_16X16X4_F32 A-matrix layout:
    //   slot i -> column col(i) = 4*(i/2) + 2*h + (i&1), row = lane&15
    float adv[64], xir[64], wt[64], g[64];

#pragma unroll
    for (int i = 0; i < 64; ++i) {
        int col = 4 * (i >> 1) + 2 * h + (i & 1);
        float xv = (col < DIN_) ? x[(size_t)myRow * DIN_ + col] : y[myRow];
        xir[i] = xv;
        adv[i] = xv;
        wt[i]  = (col < DIN_) ? w[col] : -1.0f;   // wt = [w, -1]
    }

    float lossAcc = 0.0f;

    for (int t = 0; t <= T_STEPS; ++t) {
        // ---- residual r[m] = sum_k adv[m][k] * wt[k], via 32 chained WMMAs ----
        // B = wt replicated across all 16 columns -> C[m][n] == r[m] for all n.
        v8f c = {0.f, 0.f, 0.f, 0.f, 0.f, 0.f, 0.f, 0.f};
#pragma unroll
        for (int j = 0; j < 32; ++j) {
            v2f av; av.x = adv[2 * j]; av.y = adv[2 * j + 1];
            v2f bv; bv.x = wt[2 * j];  bv.y = wt[2 * j + 1];
            c = __builtin_amdgcn_wmma_f32_16x16x4_f32(
                    false, av, false, bv, (short)0, c, false, false);
        }
        // broadcast r[row] back to both lanes owning that row:
        // s_L = C[(L&7)+(L>=16?8:0)]; lanes in [8,24) need the xor-16 partner.
        int m7 = lane & 7;
        float s = c[0];
        s = (m7 == 1) ? c[1] : s;
        s = (m7 == 2) ? c[2] : s;
        s = (m7 == 3) ? c[3] : s;
        s = (m7 == 4) ? c[4] : s;
        s = (m7 == 5) ? c[5] : s;
        s = (m7 == 6) ? c[6] : s;
        s = (m7 == 7) ? c[7] : s;
        float so = __shfl_xor(s, 16, 32);
        bool swap = (lane >= 8) && (lane < 24);
        float r = (swap ? so : s) + b0c;

        // ---- collect loss for the last M samples (adv == sample of step t-1) ----
        if (t > T_STEPS - M_KEEP) {
            float cp = 0.0f;
#pragma unroll
            for (int i = 0; i < 64; ++i) { float d = adv[i] - xir[i]; cp += d * d; }
            float cost = cp + __shfl_xor(cp, 16, 32);   // full row cost on both halves
            if (lane < 16) lossAcc += r * r - lbdc * cost;
        }
        if (t == T_STEPS) break;

        // ---- gradient of log_pi_sum and local squared norm ----
        float n2 = 0.0f;
#pragma unroll
        for (int i = 0; i < 64; ++i) {
            float gi = (2.0f * r * wt[i] - 2.0f * c2 * (adv[i] - xir[i])) * (1.0f / EPS_);
            g[i] = gi;
            n2 += gi * gi;
        }
        // wave reduce
#pragma unroll
        for (int k = 16; k >= 1; k >>= 1) n2 += __shfl_xor(n2, k, 32);
        if (lane == 0) sred[wInB] = n2;
        __syncthreads();
        if (threadIdx.x == 0) {
            float tot = 0.0f;
#pragma unroll
            for (int v = 0; v < WPB; ++v) tot += sred[v];
            normP[(t & 1) * NB + blockIdx.x] = tot;   // double-buffered by parity
        }
        grid_barrier(&bar[0], &bar[1], NB);

        // deterministic global norm: every wave sums all NB partials
        {
            const volatile float* vp = normP + (t & 1) * NB;
            float tb = vp[lane] + vp[lane + 32];
#pragma unroll
            for (int k = 16; k >= 1; k >>= 1) tb += __shfl_xor(tb, k, 32);
            float scale = BOUND_ / fmaxf(BOUND_, __builtin_sqrtf(tb));
            float lrS = LR_ * scale;
            // ---- Langevin update ----
#pragma unroll
            for (int i = 0; i < 64; ++i) {
                int col = 4 * (i >> 1) + 2 * h + (i & 1);
                unsigned id = (unsigned)myRow * 128u + (unsigned)col;
                float z = gauss_rand(id, (unsigned)t);
                adv[i] += lrS * g[i] + nsc * z;
            }
        }
    }

    // ---- final deterministic loss reduction ----
#pragma unroll
    for (int k = 16; k >= 1; k >>= 1) lossAcc += __shfl_xor(lossAcc, k, 32);
    if (lane == 0) sred[wInB] = lossAcc;
    __syncthreads();
    if (threadIdx.x == 0) {
        float tot = 0.0f;
#pragma unroll
        for (int v = 0; v < WPB; ++v) tot += sred[v];
        lossP[blockIdx.x] = tot;
    }
    grid_barrier(&bar[0], &bar[1], NB);

    if (blockIdx.x == 0 && wInB == 0) {
        const volatile float* lp = lossP;
        float v = lp[lane] + lp[lane + 32];
#pragma unroll
        for (int k = 16; k >= 1; k >>= 1) v += __shfl_xor(v, k, 32);
        if (lane == 0)
            out[0] = lbdc * RHO_ + v / (float)((size_t)B_ROWS * M_KEEP);
    }
}

extern "C" void kernel_launch(void* const* d_in, const int* in_sizes, int n_in,
                              void* d_out, int out_size, void* d_ws, size_t ws_size,
                              hipStream_t stream) {
    (void)in_sizes; (void)n_in; (void)out_size; (void)ws_size;
    const float* x   = (const float*)d_in[0];
    const float* y   = (const float*)d_in[1];
    const float* w   = (const float*)d_in[2];
    const float* b0  = (const float*)d_in[3];
    const float* lbd = (const float*)d_in[4];
    float* out = (float*)d_out;

    unsigned* bar  = (unsigned*)d_ws;                       // [0]=cnt [1]=gen
    float* normP   = (float*)((char*)d_ws + 64);            // 2*NB floats (parity buffers)
    float* lossP   = normP + 2 * NB;                        // NB floats

    // reset barrier state each call (graph-capture safe: becomes a memset node)
    hipMemsetAsync(d_ws, 0, 64, stream);

    hipLaunchKernelGGL(langevin_dro_kernel, dim3(NB), dim3(TPB), 0, stream,
                       x, y, w, b0, lbd, out, bar, normP, lossP);
}